// GCN_30485677867756
// MI455X (gfx1250) — compile-verified
//
#include <hip/hip_runtime.h>
#include <hip/hip_bf16.h>
#include <math.h>

// ---------------------------------------------------------------------------
// GCN 2-layer forward for MI455X (gfx1250, wave32).
//   h1   = x @ W1             (WMMA f32 16x16x4, exact f32, 80-row M blocks)
//   agg1 = scatter_add(h1[src] -> dst)   (f32 atomics, L2-resident)
//   h1'  = relu(agg1 + b1)
//   h2   = h1' @ W2pad        (W2 zero-padded 128x40 -> 128x48, guard-free)
//   agg2 = scatter_add(h2[src] -> dst)
//   out  = log_softmax(agg2 + b2)
// ---------------------------------------------------------------------------

typedef __attribute__((ext_vector_type(2))) float v2f;
typedef __attribute__((ext_vector_type(8))) float v8f;

#define GCN_NODES 50000
#define GCN_EDGES 800000
#define GCN_K     128     // inner dim for both GEMMs
#define A_PITCH   132     // 128 + 4 pad: conflict-free LDS row reads

// ---------------------------------------------------------------------------
// Zero fill (float4 granularity)
// ---------------------------------------------------------------------------
__global__ void __launch_bounds__(256)
gcn_zero_f32(float* __restrict__ p, int n4)
{
    const int tid = blockIdx.x * 256 + threadIdx.x;
    if (tid < n4) ((float4*)p)[tid] = make_float4(0.f, 0.f, 0.f, 0.f);
}

// ---------------------------------------------------------------------------
// Zero-pad W2 [128x40] -> Bpad [128x48] so GEMM2 needs no column guard.
// ---------------------------------------------------------------------------
__global__ void __launch_bounds__(256)
gcn_pad_w2(const float* __restrict__ W2, float* __restrict__ Bpad)
{
    const int i = blockIdx.x * 256 + threadIdx.x;
    if (i >= 128 * 48) return;
    const int r = i / 48;
    const int c = i - r * 48;
    Bpad[i] = (c < 40) ? W2[r * 40 + c] : 0.f;
}

// ---------------------------------------------------------------------------
// WMMA GEMM: C[M x NOUT] = A[M x 128] * B[128 x NP]  (NP >= NOUT, B pre-padded)
//   grid.x = M / (MT*16), block = (32, NTILES): one wave per 16-col N-tile.
//   Each block stages MT*16 rows of A (contiguous) into LDS; each wave keeps
//   MT accumulators (one per M-subtile) so every B fetch feeds MT WMMAs.
//   NP is a compile-time constant -> all B loads are base+imm global_load_b32.
//   EXEC is all-ones around every v_wmma (no divergence in the k-loop).
// ---------------------------------------------------------------------------
template<int NTILES, int NOUT, int NP, int MT>
__global__ void __launch_bounds__(32 * NTILES)
gcn_gemm_wmma(const float* __restrict__ A, const float* __restrict__ B,
              float* __restrict__ C)
{
    __shared__ float As[MT * 16 * A_PITCH];

    constexpr int NTHR = 32 * NTILES;
    const int t  = threadIdx.y * 32 + threadIdx.x;
    const int m0 = blockIdx.x * (MT * 16);

    // Stage A: MT*16 rows x 128 floats, contiguous in memory (b128 loads)
    {
        const float4* __restrict__ asrc =
            (const float4*)(A + (size_t)m0 * GCN_K);
        for (int i = t; i < MT * 16 * 32; i += NTHR) {
            const int row = i >> 5;
            const int c4  = i & 31;
            const float4 v = asrc[i];
            float* d = &As[row * A_PITCH + (c4 << 2)];
            d[0] = v.x; d[1] = v.y; d[2] = v.z; d[3] = v.w;
        }
    }
    __syncthreads();

    const int lane = threadIdx.x;                 // wave32 lane
    const int w    = threadIdx.y;                 // n-tile index
    const int col  = (w << 4) + (lane & 15);      // column (padded space)
    const int koff = (lane >> 4) << 1;            // lanes 0-15 -> K{0,1}, 16-31 -> K{2,3}
    const int arow = lane & 15;

    v8f acc[MT];
    #pragma unroll
    for (int mt = 0; mt < MT; ++mt)
        acc[mt] = (v8f){0.f, 0.f, 0.f, 0.f, 0.f, 0.f, 0.f, 0.f};

    // Per-lane B base pointer; all k offsets become immediates after unroll.
    const float* __restrict__ bp = B + koff * NP + col;

    #pragma unroll
    for (int k = 0; k < GCN_K; k += 4) {
        v2f b;
        b.x = bp[k * NP];             // B[k+koff  ][col]
        b.y = bp[(k + 1) * NP];       // B[k+koff+1][col]
        #pragma unroll
        for (int mt = 0; mt < MT; ++mt) {
            const float* ap = &As[(mt * 16 + arow) * A_PITCH + k + koff];
            v2f a;
            a.x = ap[0];
            a.y = ap[1];
            acc[mt] = __builtin_amdgcn_wmma_f32_16x16x4_f32(
                /*neg_a=*/false, a, /*neg_b=*/false, b,
                /*c_mod=*/(short)0, acc[mt], /*reuse_a=*/false, /*reuse_b=*/false);
        }
    }

    // C/D layout: VGPR r -> row (r + 8*(lane>=16)), column fixed per lane.
    if (col < NOUT) {
        #pragma unroll
        for (int mt = 0; mt < MT; ++mt) {
            const int mbase = m0 + mt * 16 + ((lane >> 4) << 3);
            #pragma unroll
            for (int r = 0; r < 8; ++r)
                C[(size_t)(mbase + r) * NOUT + col] = acc[mt][r];
        }
    }
}

// ---------------------------------------------------------------------------
// Edge scatter-add: agg[dst[e]] += h[src[e]], C = 4*C4 channels.
// Thread handles one (edge, float4-chunk). agg (<=25.6 MB) is L2-resident,
// so the f32 atomics resolve at L2 bandwidth. C4 is compile-time.
// ---------------------------------------------------------------------------
template<int C4>
__global__ void __launch_bounds__(256)
gcn_scatter_add(const long long* __restrict__ eidx, const float* __restrict__ h,
                float* __restrict__ agg)
{
    const int tid = blockIdx.x * 256 + threadIdx.x;
    if (tid >= GCN_EDGES * C4) return;
    const int e = tid / C4;            // constant divisor -> mul/shift
    const int q = tid - e * C4;
    const int s = (int)eidx[e];
    const int d = (int)eidx[GCN_EDGES + e];
    const float4 v = ((const float4*)(h + (size_t)s * (C4 * 4)))[q];
    float* o = agg + (size_t)d * (C4 * 4) + (q << 2);
    atomicAdd(o + 0, v.x);
    atomicAdd(o + 1, v.y);
    atomicAdd(o + 2, v.z);
    atomicAdd(o + 3, v.w);
}

// ---------------------------------------------------------------------------
// out = relu(agg + bias), 128 channels, float4 vectorized
// ---------------------------------------------------------------------------
__global__ void __launch_bounds__(256)
gcn_bias_relu(const float* __restrict__ agg, const float* __restrict__ bias,
              float* __restrict__ out)
{
    const int tid = blockIdx.x * 256 + threadIdx.x;
    if (tid >= GCN_NODES * 32) return;
    const float4 v = ((const float4*)agg)[tid];
    const float4 b = ((const float4*)bias)[tid & 31];
    float4 r;
    r.x = fmaxf(v.x + b.x, 0.f);
    r.y = fmaxf(v.y + b.y, 0.f);
    r.z = fmaxf(v.z + b.z, 0.f);
    r.w = fmaxf(v.w + b.w, 0.f);
    ((float4*)out)[tid] = r;
}

// ---------------------------------------------------------------------------
// out = log_softmax(agg + bias) over 40 channels; one wave32 per row.
// Lane holds c=lane (all 32) and c=lane+32 (lanes 0..7); xor-shuffle reduce.
// ---------------------------------------------------------------------------
__global__ void __launch_bounds__(256)
gcn_bias_logsoftmax(const float* __restrict__ agg, const float* __restrict__ bias,
                    float* __restrict__ out)
{
    const int wid  = (blockIdx.x * 256 + threadIdx.x) >> 5;
    const int lane = threadIdx.x & 31;
    if (wid >= GCN_NODES) return;
    const float* r = agg + (size_t)wid * 40;
    const float v0 = r[lane] + bias[lane];
    const float v1 = (lane < 8) ? (r[32 + lane] + bias[32 + lane]) : -INFINITY;

    float m = fmaxf(v0, v1);
    #pragma unroll
    for (int o = 16; o > 0; o >>= 1) m = fmaxf(m, __shfl_xor(m, o, 32));

    float s = expf(v0 - m) + ((lane < 8) ? expf(v1 - m) : 0.f);
    #pragma unroll
    for (int o = 16; o > 0; o >>= 1) s += __shfl_xor(s, o, 32);

    const float lse = m + logf(s);
    out[(size_t)wid * 40 + lane] = v0 - lse;
    if (lane < 8) out[(size_t)wid * 40 + 32 + lane] = v1 - lse;
}

// ---------------------------------------------------------------------------
// Launcher
// ---------------------------------------------------------------------------
extern "C" void kernel_launch(void* const* d_in, const int* in_sizes, int n_in,
                              void* d_out, int out_size, void* d_ws, size_t ws_size,
                              hipStream_t stream)
{
    (void)in_sizes; (void)n_in; (void)out_size; (void)ws_size;

    const float*     x  = (const float*)d_in[0];      // [50000,128]
    const long long* ei = (const long long*)d_in[1];  // [2,800000] int64
    const float*     W1 = (const float*)d_in[2];      // [128,128]
    const float*     b1 = (const float*)d_in[3];      // [128]
    const float*     W2 = (const float*)d_in[4];      // [128,40]
    const float*     b2 = (const float*)d_in[5];      // [40]
    float*           out = (float*)d_out;             // [50000,40]

    // Workspace layout (all f32): h1 | agg1 | h2 | agg2 | W2pad  (~67.2 MB)
    float* h1    = (float*)d_ws;                       // 50000*128
    float* agg1  = h1    + (size_t)GCN_NODES * 128;    // 50000*128
    float* h2    = agg1  + (size_t)GCN_NODES * 128;    // 50000*40
    float* agg2  = h2    + (size_t)GCN_NODES * 40;     // 50000*40
    float* w2pad = agg2  + (size_t)GCN_NODES * 40;     // 128*48
    // h1' (post-ReLU) reuses the h1 buffer.

    // 0. zero accumulators + pad W2 (independent of GEMM1)
    {
        const int n4a = GCN_NODES * 128 / 4;
        const int n4b = GCN_NODES * 40 / 4;
        gcn_zero_f32<<<(n4a + 255) / 256, 256, 0, stream>>>(agg1, n4a);
        gcn_zero_f32<<<(n4b + 255) / 256, 256, 0, stream>>>(agg2, n4b);
        gcn_pad_w2<<<(128 * 48 + 255) / 256, 256, 0, stream>>>(W2, w2pad);
    }

    // 1. h1 = x @ W1    (625 blocks x 80 rows, 8 waves = 8 N-tiles)
    gcn_gemm_wmma<8, 128, 128, 5>
        <<<GCN_NODES / 80, dim3(32, 8), 0, stream>>>(x, W1, h1);

    // 2. agg1 = scatter_add(h1)
    gcn_scatter_add<32>
        <<<(GCN_EDGES * 32 + 255) / 256, 256, 0, stream>>>(ei, h1, agg1);

    // 3. h1' = relu(agg1 + b1)
    gcn_bias_relu<<<(GCN_NODES * 32 + 255) / 256, 256, 0, stream>>>(agg1, b1, h1);

    // 4. h2 = h1' @ W2pad   (3 waves = 3 N-tiles, cols 40..47 are zero-padded)
    gcn_gemm_wmma<3, 40, 48, 5>
        <<<GCN_NODES / 80, dim3(32, 3), 0, stream>>>(h1, w2pad, h2);

    // 5. agg2 = scatter_add(h2)
    gcn_scatter_add<10>
        <<<(GCN_EDGES * 10 + 255) / 256, 256, 0, stream>>>(ei, h2, agg2);

    // 6. out = log_softmax(agg2 + b2)   (8 rows per 256-thread block)
    gcn_bias_logsoftmax<<<(GCN_NODES + 7) / 8, 256, 0, stream>>>(agg2, b2, out);
}